// GAT_81801947120299
// MI455X (gfx1250) — compile-verified
//
#include <hip/hip_runtime.h>
#include <hip/hip_bf16.h>
#include <stdint.h>

// ---------------- problem constants (from reference) ----------------
#define N_NODES 10000
#define E_EDGES 160000
#define IN_CH   128
#define HEADS1  8
#define C1_CH   128
#define F1      (HEADS1 * C1_CH)   // 1024
#define OUT_CH  64
#define NEG_SLOPE 0.2f

typedef __attribute__((ext_vector_type(2))) float v2f;
typedef __attribute__((ext_vector_type(8))) float v8f;
typedef __attribute__((ext_vector_type(4))) int   v4i;

typedef __attribute__((address_space(1))) v4i gv4i_t;   // global int4
typedef __attribute__((address_space(3))) v4i lv4i_t;   // LDS int4

#if defined(__has_builtin)
#if __has_builtin(__builtin_amdgcn_global_load_async_to_lds_b128)
#define USE_ASYNC_LDS 1
#endif
#endif

__device__ __forceinline__ void wait_asynccnt0() {
#if defined(__has_builtin) && __has_builtin(__builtin_amdgcn_s_wait_asynccnt)
    __builtin_amdgcn_s_wait_asynccnt(0);
#else
    asm volatile("s_wait_asynccnt 0x0" ::: "memory");
#endif
}

// ---------------- fp32 WMMA GEMM with LDS-staged B panel ----------------
// C[M,Nn] = A[M,K] @ B[K,Nn].  Requires: M % 16 == 0, Nn % 64 == 0, K % KC == 0.
// Block = 256 threads = 8 waves. Block computes a 128-row x 64-col output block:
// each wave owns a 16x64 tile (4 accumulators), sharing one LDS-staged B panel.
#define KC 128   // K-chunk staged in LDS: 128 x 64 floats = 32 KB

__global__ void __launch_bounds__(256)
gemm_wmma_f32_lds(const float* __restrict__ A,
                  const float* __restrict__ B,
                  float* __restrict__ C,
                  int M, int Nn, int K) {
    __shared__ float Bs[KC * 64];

    const int wv   = threadIdx.x >> 5;
    const int lane = threadIdx.x & 31;
    const int tid  = threadIdx.x;

    const int tilesM  = M >> 4;
    const int nPanels = Nn >> 6;                 // 64-col panels
    const int mblk = blockIdx.x / nPanels;
    const int pn   = blockIdx.x - mblk * nPanels;
    const int n0   = pn << 6;

    const int  mt    = mblk * 8 + wv;            // this wave's 16-row tile index
    const bool valid = (mt < tilesM);            // wave-uniform predicate

    const int ml = lane & 15;                    // A row in tile / B,C column in 16-group
    const int kh = lane >> 4;                    // half-select

    v8f acc0 = {}, acc1 = {}, acc2 = {}, acc3 = {};

    // never dereferenced when !valid
    const float* __restrict__ Arow = A + (size_t)(mt * 16 + ml) * K;

    for (int kc = 0; kc < K; kc += KC) {
        // ---- cooperative stage of B[kc:kc+KC, n0:n0+64] into LDS ----
#pragma unroll
        for (int i = 0; i < (KC * 64 / 4) / 256; ++i) {   // 8 float4s per thread
            const int f = tid + i * 256;                  // float4 index in panel
            const int k = f >> 4;                         // panel row (16 float4 per row)
            const int c = (f & 15) << 2;                  // col within 64
            const float* gp = B + (size_t)(kc + k) * Nn + (n0 + c);
#if USE_ASYNC_LDS
            __builtin_amdgcn_global_load_async_to_lds_b128(
                (gv4i_t*)gp, (lv4i_t*)&Bs[f << 2], 0, 0);
#else
            *(float4*)&Bs[f << 2] = *(const float4*)gp;
#endif
        }
#if USE_ASYNC_LDS
        wait_asynccnt0();
#endif
        __syncthreads();

        if (valid) {
            for (int k = 0; k < KC; k += 4) {
                __builtin_prefetch(&Arow[kc + k + 64], 0, 3);
                v2f a;
                a.x = Arow[kc + k + 2 * kh];
                a.y = Arow[kc + k + 2 * kh + 1];
                const float* brow0 = &Bs[(k + 2 * kh) * 64 + ml];
                const float* brow1 = &Bs[(k + 2 * kh + 1) * 64 + ml];
                v2f b0, b1, b2, b3;
                b0.x = brow0[0];  b0.y = brow1[0];
                b1.x = brow0[16]; b1.y = brow1[16];
                b2.x = brow0[32]; b2.y = brow1[32];
                b3.x = brow0[48]; b3.y = brow1[48];
                acc0 = __builtin_amdgcn_wmma_f32_16x16x4_f32(false, a, false, b0, (short)0, acc0, false, false);
                acc1 = __builtin_amdgcn_wmma_f32_16x16x4_f32(false, a, false, b1, (short)0, acc1, false, false);
                acc2 = __builtin_amdgcn_wmma_f32_16x16x4_f32(false, a, false, b2, (short)0, acc2, false, false);
                acc3 = __builtin_amdgcn_wmma_f32_16x16x4_f32(false, a, false, b3, (short)0, acc3, false, false);
            }
        }
        __syncthreads();
    }

    if (valid) {
        // C/D layout: VGPR r -> M = r (lanes 0-15) / r+8 (lanes 16-31), N = lane%16
        float* __restrict__ Cp = C + (size_t)(mt * 16 + kh * 8) * Nn + (n0 + ml);
#pragma unroll
        for (int r = 0; r < 8; ++r) {
            Cp[(size_t)r * Nn + 0]  = acc0[r];
            Cp[(size_t)r * Nn + 16] = acc1[r];
            Cp[(size_t)r * Nn + 32] = acc2[r];
            Cp[(size_t)r * Nn + 48] = acc3[r];
        }
    }
}

// ---------------- per-(node,head) attention logits ----------------
__global__ void alpha_kernel(const float* __restrict__ h,
                             const float* __restrict__ a_src,
                             const float* __restrict__ a_dst,
                             float* __restrict__ as_out,
                             float* __restrict__ ad_out,
                             int n_nodes, int H, int C) {
    const int wave = blockIdx.x * (blockDim.x >> 5) + (threadIdx.x >> 5);
    const int lane = threadIdx.x & 31;
    if (wave >= n_nodes * H) return;
    const int node = wave / H;
    const int hd   = wave - node * H;

    const float* __restrict__ hp = h + ((size_t)node * H + hd) * C;
    const float* __restrict__ sv = a_src + (size_t)hd * C;
    const float* __restrict__ dv = a_dst + (size_t)hd * C;

    float ss = 0.f, sd = 0.f;
    for (int c = lane; c < C; c += 32) {
        float v = hp[c];
        ss += v * sv[c];
        sd += v * dv[c];
    }
#pragma unroll
    for (int off = 16; off > 0; off >>= 1) {
        ss += __shfl_xor(ss, off, 32);
        sd += __shfl_xor(sd, off, 32);
    }
    if (lane == 0) { as_out[wave] = ss; ad_out[wave] = sd; }
}

// ---------------- helpers ----------------
__device__ __forceinline__ void atomicMaxF(float* addr, float val) {
    if (val >= 0.f) atomicMax((int*)addr, __float_as_int(val));
    else            atomicMin((unsigned int*)addr, __float_as_uint(val));
}

__device__ __forceinline__ float leaky(float v) {
    return v > 0.f ? v : NEG_SLOPE * v;
}

__device__ __forceinline__ void edge_endpoints(const long long* __restrict__ ei,
                                               int e, int& s, int& d) {
    if (e < E_EDGES) { s = (int)ei[e]; d = (int)ei[E_EDGES + e]; }
    else             { s = d = e - E_EDGES; }   // appended self-loops
}

// ---------------- edge pass 1: segment max ----------------
__global__ void edge_max_kernel(const long long* __restrict__ ei, int EP, int H,
                                const float* __restrict__ as_in,
                                const float* __restrict__ ad_in,
                                float* __restrict__ m) {
    const int t = blockIdx.x * blockDim.x + threadIdx.x;
    if (t >= EP * H) return;
    const int e = t / H, hd = t - e * H;
    int s, d; edge_endpoints(ei, e, s, d);
    const float v = leaky(as_in[s * H + hd] + ad_in[d * H + hd]);
    atomicMaxF(&m[d * H + hd], v);
}

// ---------------- edge pass 2: segment sum of exp ----------------
__global__ void edge_sum_kernel(const long long* __restrict__ ei, int EP, int H,
                                const float* __restrict__ as_in,
                                const float* __restrict__ ad_in,
                                const float* __restrict__ m,
                                float* __restrict__ sum) {
    const int t = blockIdx.x * blockDim.x + threadIdx.x;
    if (t >= EP * H) return;
    const int e = t / H, hd = t - e * H;
    int s, d; edge_endpoints(ei, e, s, d);
    const float v = leaky(as_in[s * H + hd] + ad_in[d * H + hd]);
    atomicAdd(&sum[d * H + hd], __expf(v - m[d * H + hd]));
}

// ---------------- edge pass 3: weighted message aggregation ----------------
__global__ void edge_agg_kernel(const long long* __restrict__ ei, int EP, int H, int C,
                                const float* __restrict__ as_in,
                                const float* __restrict__ ad_in,
                                const float* __restrict__ m,
                                const float* __restrict__ sum,
                                const float* __restrict__ h,
                                float* __restrict__ out) {
    const int t = blockIdx.x * blockDim.x + threadIdx.x;
    if (t >= EP * H) return;
    const int e = t / H, hd = t - e * H;
    int s, d; edge_endpoints(ei, e, s, d);
    const float v = leaky(as_in[s * H + hd] + ad_in[d * H + hd]);
    const float alpha = __expf(v - m[d * H + hd]) / (sum[d * H + hd] + 1e-16f);

    const float4* __restrict__ hp = (const float4*)(h + ((size_t)s * H + hd) * C);
    float* __restrict__ op = out + ((size_t)d * H + hd) * C;
    const int c4n = C >> 2;
    for (int c4 = 0; c4 < c4n; ++c4) {
        float4 hv = hp[c4];
        atomicAdd(op + c4 * 4 + 0, alpha * hv.x);
        atomicAdd(op + c4 * 4 + 1, alpha * hv.y);
        atomicAdd(op + c4 * 4 + 2, alpha * hv.z);
        atomicAdd(op + c4 * 4 + 3, alpha * hv.w);
    }
}

// ---------------- bias (+ optional ELU) epilogue, in place ----------------
__global__ void bias_act_kernel(float* __restrict__ g, const float* __restrict__ b,
                                int n_nodes, int F, int do_elu) {
    const size_t t = (size_t)blockIdx.x * blockDim.x + threadIdx.x;
    if (t >= (size_t)n_nodes * F) return;
    const int col = (int)(t % F);
    float v = g[t] + b[col];
    if (do_elu) v = v > 0.f ? v : (__expf(v) - 1.f);
    g[t] = v;
}

// ---------------- constant fill (for -inf init of segment max) ----------------
__global__ void fill_kernel(float* __restrict__ p, float v, size_t n) {
    const size_t t = (size_t)blockIdx.x * blockDim.x + threadIdx.x;
    if (t < n) p[t] = v;
}

// ---------------- host driver ----------------
extern "C" void kernel_launch(void* const* d_in, const int* in_sizes, int n_in,
                              void* d_out, int out_size, void* d_ws, size_t ws_size,
                              hipStream_t stream) {
    const float*     x      = (const float*)d_in[0];
    const long long* ei     = (const long long*)d_in[1];  // int64 [2,E]
    const float*     W1     = (const float*)d_in[2];
    const float*     a_src1 = (const float*)d_in[3];
    const float*     a_dst1 = (const float*)d_in[4];
    const float*     b1     = (const float*)d_in[5];
    const float*     W2     = (const float*)d_in[6];
    const float*     a_src2 = (const float*)d_in[7];
    const float*     a_dst2 = (const float*)d_in[8];
    const float*     b2     = (const float*)d_in[9];
    float*           out    = (float*)d_out;             // [N, 64]

    (void)in_sizes; (void)n_in; (void)out_size; (void)ws_size;

    // ---- workspace layout (floats) ----
    float* ws = (float*)d_ws;
    size_t o = 0;
    float* h1  = ws + o; o += (size_t)N_NODES * F1;        // x@W1        [N,8,128]
    float* g1  = ws + o; o += (size_t)N_NODES * F1;        // layer1 aggregate / layer2 input
    float* as1 = ws + o; o += (size_t)N_NODES * HEADS1;
    float* ad1 = ws + o; o += (size_t)N_NODES * HEADS1;
    float* m1  = ws + o; o += (size_t)N_NODES * HEADS1;
    float* s1  = ws + o; o += (size_t)N_NODES * HEADS1;
    float* z2  = ws + o; o += (size_t)N_NODES * OUT_CH;    // g1@W2       [N,64]
    float* as2 = ws + o; o += (size_t)N_NODES;
    float* ad2 = ws + o; o += (size_t)N_NODES;
    float* m2  = ws + o; o += (size_t)N_NODES;
    float* s2  = ws + o; o += (size_t)N_NODES;

    const int EP = E_EDGES + N_NODES;                      // edges incl. self-loops

    // ---- init ----
    (void)hipMemsetAsync(g1, 0, sizeof(float) * (size_t)N_NODES * F1, stream);
    (void)hipMemsetAsync(s1, 0, sizeof(float) * (size_t)N_NODES * HEADS1, stream);
    (void)hipMemsetAsync(s2, 0, sizeof(float) * (size_t)N_NODES, stream);
    (void)hipMemsetAsync(out, 0, sizeof(float) * (size_t)N_NODES * OUT_CH, stream);
    {
        size_t n = (size_t)N_NODES * HEADS1;
        fill_kernel<<<(unsigned)((n + 255) / 256), 256, 0, stream>>>(m1, -INFINITY, n);
        fill_kernel<<<(unsigned)((N_NODES + 255) / 256), 256, 0, stream>>>(m2, -INFINITY, (size_t)N_NODES);
    }

    const int tilesM = N_NODES / 16;            // 625
    const int mblks  = (tilesM + 7) / 8;        // 79

    // ---- layer 1 ----
    {   // h1 = x @ W1  (10000x128 @ 128x1024): 79 m-blocks x 16 n-panels
        const int nPanels = F1 / 64;            // 16
        gemm_wmma_f32_lds<<<mblks * nPanels, 256, 0, stream>>>(x, W1, h1, N_NODES, F1, IN_CH);
    }
    {
        const int waves = N_NODES * HEADS1;
        alpha_kernel<<<(waves + 7) / 8, 256, 0, stream>>>(h1, a_src1, a_dst1, as1, ad1,
                                                          N_NODES, HEADS1, C1_CH);
    }
    {
        const int tot = EP * HEADS1;
        const unsigned g = (tot + 255) / 256;
        edge_max_kernel<<<g, 256, 0, stream>>>(ei, EP, HEADS1, as1, ad1, m1);
        edge_sum_kernel<<<g, 256, 0, stream>>>(ei, EP, HEADS1, as1, ad1, m1, s1);
        edge_agg_kernel<<<g, 256, 0, stream>>>(ei, EP, HEADS1, C1_CH, as1, ad1, m1, s1, h1, g1);
    }
    {   // g1 = elu(g1 + b1)
        const size_t n = (size_t)N_NODES * F1;
        bias_act_kernel<<<(unsigned)((n + 255) / 256), 256, 0, stream>>>(g1, b1, N_NODES, F1, 1);
    }

    // ---- layer 2 ----
    {   // z2 = g1 @ W2  (10000x1024 @ 1024x64): 79 m-blocks x 1 n-panel
        gemm_wmma_f32_lds<<<mblks, 256, 0, stream>>>(g1, W2, z2, N_NODES, OUT_CH, F1);
    }
    {
        alpha_kernel<<<(N_NODES + 7) / 8, 256, 0, stream>>>(z2, a_src2, a_dst2, as2, ad2,
                                                            N_NODES, 1, OUT_CH);
    }
    {
        const unsigned g = (EP + 255) / 256;
        edge_max_kernel<<<g, 256, 0, stream>>>(ei, EP, 1, as2, ad2, m2);
        edge_sum_kernel<<<g, 256, 0, stream>>>(ei, EP, 1, as2, ad2, m2, s2);
        edge_agg_kernel<<<g, 256, 0, stream>>>(ei, EP, 1, OUT_CH, as2, ad2, m2, s2, z2, out);
    }
    {   // out += b2
        const size_t n = (size_t)N_NODES * OUT_CH;
        bias_act_kernel<<<(unsigned)((n + 255) / 256), 256, 0, stream>>>(out, b2, N_NODES, OUT_CH, 0);
    }
}